// DecoderBase_10015863734734
// MI455X (gfx1250) — compile-verified
//
#include <hip/hip_runtime.h>
#include <stdint.h>

#define TT 32
#define BB 16
#define KK 10
#define VV 50257
#define BK (BB*KK)
#define LPEN 0.7f
#define NEGBIG (-1.0e8f)
#define SENT_IDX 0x7FFFFFFF
#define TPB_A 512
#define TILE 2048      // TPB_A * 4 floats = 8KB
#define SEGS 4
#define SEGLEN 12565   // ceil(VV/SEGS); last segment is 12562
#define NT 6           // full 2048-float tiles per segment after alignment pad (always 6)

// ---------------- workspace carving ----------------
struct WS {
  int*      flagA;     // [BK]
  float*    score;     // [BK]
  float*    len;       // [BK]
  unsigned* minbits;   // [TT]   per-step global-min (monotone uint encoding)
  int*      rec_wid;   // [TT*BK]
  int*      rec_rg;    // [TT*BK]
  int*      rec_em;    // [TT*BK]
  float*    part_val;  // [BK*10] per-row rank values (new_score)
  float*    part_cand; // [BK*10] per-row cand values (sum_score)
  int*      part_idx;  // [BK*10] per-row flat k*V+v
  float*    seg_max;   // [BK*SEGS]
  float*    seg_min;   // [BK*SEGS]
  float*    seg_sum;   // [BK*SEGS]  sum exp(x - segmax)
  float*    seg_tv;    // [BK*SEGS*10] raw top logits (desc, idx asc)
  int*      seg_ti;    // [BK*SEGS*10] absolute v within row
};
__host__ __device__ static inline WS carve(void* ws) {
  WS w; char* p = (char*)ws;
  w.flagA    = (int*)p;      p += BK*4;
  w.score    = (float*)p;    p += BK*4;
  w.len      = (float*)p;    p += BK*4;
  w.minbits  = (unsigned*)p; p += TT*4;
  w.rec_wid  = (int*)p;      p += TT*BK*4;
  w.rec_rg   = (int*)p;      p += TT*BK*4;
  w.rec_em   = (int*)p;      p += TT*BK*4;
  w.part_val = (float*)p;    p += BK*10*4;
  w.part_cand= (float*)p;    p += BK*10*4;
  w.part_idx = (int*)p;      p += BK*10*4;
  w.seg_max  = (float*)p;    p += BK*SEGS*4;
  w.seg_min  = (float*)p;    p += BK*SEGS*4;
  w.seg_sum  = (float*)p;    p += BK*SEGS*4;
  w.seg_tv   = (float*)p;    p += BK*SEGS*10*4;
  w.seg_ti   = (int*)p;      p += BK*SEGS*10*4;
  return w;
}

__device__ __forceinline__ unsigned f2mono(float x) {
  unsigned u = __float_as_uint(x);
  return (u & 0x80000000u) ? ~u : (u | 0x80000000u);
}
__device__ __forceinline__ float mono2f(unsigned u) {
  unsigned b = (u & 0x80000000u) ? (u ^ 0x80000000u) : ~u;
  return __uint_as_float(b);
}
__device__ __forceinline__ bool better(float v1, int i1, float v2, int i2) {
  return (v1 > v2) || ((v1 == v2) && (i1 < i2));
}

// ---------------- CDNA5 async global->LDS (ASYNCcnt path) ----------------
#if defined(__AMDGCN__)
__device__ __forceinline__ void async_ld16(unsigned ldsoff, const float* g) {
  // per-lane: copies 16 bytes from global address g to LDS address ldsoff, tracked by ASYNCcnt
  asm volatile("global_load_async_to_lds_b128 %0, %1, off"
               :: "v"(ldsoff), "v"(g) : "memory");
}
// asm (not builtin) so the "memory" clobber orders subsequent ds_loads after the wait
#define WAIT_ASYNC(n) asm volatile("s_wait_asynccnt %0" :: "i"(n) : "memory")
#endif

// ---------------- setup ----------------
__global__ void k_setup(void* wsraw) {
  WS w = carve(wsraw);
  int tid = threadIdx.x;
  if (tid < BK) {
    w.flagA[tid] = 0;
    w.score[tid] = ((tid % KK) == 0) ? 0.0f : NEGBIG;
    w.len[tid]   = 0.0f;
  }
  if (tid < TT) w.minbits[tid] = 0xFFFFFFFFu;
}

// ---- kernel A1: per-(row,segment) stats + local top-10, single HBM pass ----
__global__ __launch_bounds__(TPB_A)
void k_seg(const float* __restrict__ logits, void* wsraw, int t) {
  WS w = carve(wsraw);
  const int bx = blockIdx.x;
  const int r = bx / SEGS;           // row = b*K + k
  const int sseg = bx % SEGS;
  const int tid = threadIdx.x;
  if (w.flagA[r]) return;            // finished rows handled entirely in k_rowfin

  const size_t elem0 = ((size_t)t * BK + (size_t)r) * (size_t)VV;
  const float* row = logits + elem0;
  const int S0 = sseg * SEGLEN;
  const int SE = (sseg == SEGS - 1) ? VV : (S0 + SEGLEN);
  const int pad = (int)((4u - (unsigned)((elem0 + (size_t)S0) & 3u)) & 3u);
  const int A0 = S0 + pad;           // 16B-aligned start

  __shared__ __align__(16) float tiles[NT][TILE];

  // deep async pipeline: issue the whole segment (6 x 512B per wave in flight)
#if defined(__AMDGCN__)
#pragma unroll
  for (int it = 0; it < NT; ++it)
    async_ld16((unsigned)(size_t)&tiles[it][tid * 4],
               row + A0 + it * TILE + tid * 4);
#else
  for (int it = 0; it < NT; ++it) {
    const int o = A0 + it * TILE + tid * 4;
    tiles[it][tid*4+0] = row[o+0]; tiles[it][tid*4+1] = row[o+1];
    tiles[it][tid*4+2] = row[o+2]; tiles[it][tid*4+3] = row[o+3];
  }
#endif

  float mx = -__builtin_inff(), mn = __builtin_inff();
  float lv[10]; int li[10];
#pragma unroll
  for (int q = 0; q < 10; ++q) { lv[q] = -__builtin_inff(); li[q] = SENT_IDX; }

  auto proc = [&](float x, int v) {
    mx = fmaxf(mx, x); mn = fminf(mn, x);
    if (x > lv[9]) {                 // strict: equal value keeps earlier index
      lv[9] = x; li[9] = v;
#pragma unroll
      for (int q = 9; q > 0; --q) {
        if (lv[q] > lv[q - 1]) {
          float tv = lv[q]; lv[q] = lv[q - 1]; lv[q - 1] = tv;
          int   ti = li[q]; li[q] = li[q - 1]; li[q - 1] = ti;
        }
      }
    }
  };

  // prefix scalars (smaller indices processed first per thread)
  if (tid < pad) proc(row[S0 + tid], S0 + tid);

  auto tile1 = [&](int it) {
    const float4 x4 = *(const float4*)&tiles[it][tid * 4];
    const int vb = A0 + it * TILE + tid * 4;
    proc(x4.x, vb); proc(x4.y, vb + 1); proc(x4.z, vb + 2); proc(x4.w, vb + 3);
  };
#if defined(__AMDGCN__)
  WAIT_ASYNC(5); tile1(0);
  WAIT_ASYNC(4); tile1(1);
  WAIT_ASYNC(3); tile1(2);
  WAIT_ASYNC(2); tile1(3);
  WAIT_ASYNC(1); tile1(4);
  WAIT_ASYNC(0); tile1(5);
#else
  for (int it = 0; it < NT; ++it) tile1(it);
#endif
  for (int v = A0 + NT * TILE + tid; v < SE; v += TPB_A) proc(row[v], v);

  __shared__ float s_red[TPB_A];
  __shared__ int   s_idx[TPB_A];

  s_red[tid] = mx; __syncthreads();
  for (int s = TPB_A / 2; s > 0; s >>= 1) {
    if (tid < s) s_red[tid] = fmaxf(s_red[tid], s_red[tid + s]);
    __syncthreads();
  }
  const float segmax = s_red[0]; __syncthreads();
  s_red[tid] = mn; __syncthreads();
  for (int s = TPB_A / 2; s > 0; s >>= 1) {
    if (tid < s) s_red[tid] = fminf(s_red[tid], s_red[tid + s]);
    __syncthreads();
  }
  const float segmin = s_red[0]; __syncthreads();

  // pass 2: sum exp(x - segmax) -- reads the LDS-resident segment, not HBM
  float ssum = 0.0f;
  if (tid < pad) ssum += expf(row[S0 + tid] - segmax);
#pragma unroll
  for (int it = 0; it < NT; ++it) {
    const float4 x4 = *(const float4*)&tiles[it][tid * 4];
    ssum += expf(x4.x - segmax) + expf(x4.y - segmax)
          + expf(x4.z - segmax) + expf(x4.w - segmax);
  }
  for (int v = A0 + NT * TILE + tid; v < SE; v += TPB_A)
    ssum += expf(row[v] - segmax);
  s_red[tid] = ssum; __syncthreads();
  for (int s = TPB_A / 2; s > 0; s >>= 1) {
    if (tid < s) s_red[tid] += s_red[tid + s];
    __syncthreads();
  }
  if (tid == 0) {
    w.seg_max[bx] = segmax; w.seg_min[bx] = segmin; w.seg_sum[bx] = s_red[0];
  }
  __syncthreads();

  // block-wide top-10 (raw logits): 10 rounds of (val desc, idx asc) argmax
  for (int j = 0; j < 10; ++j) {
    s_red[tid] = lv[0]; s_idx[tid] = li[0]; __syncthreads();
    for (int s = TPB_A / 2; s > 0; s >>= 1) {
      if (tid < s && better(s_red[tid + s], s_idx[tid + s], s_red[tid], s_idx[tid])) {
        s_red[tid] = s_red[tid + s]; s_idx[tid] = s_idx[tid + s];
      }
      __syncthreads();
    }
    const float bx_v = s_red[0]; const int bv = s_idx[0];
    __syncthreads();
    if (li[0] == bv) {             // unique winner advances
#pragma unroll
      for (int q = 0; q < 9; ++q) { lv[q] = lv[q + 1]; li[q] = li[q + 1]; }
      lv[9] = -__builtin_inff(); li[9] = SENT_IDX;
    }
    if (tid == 0) { w.seg_tv[bx * 10 + j] = bx_v; w.seg_ti[bx * 10 + j] = bv; }
    __syncthreads();
  }
}

// ---- kernel A2: per-row combine (logsumexp, global min, 4-way top-10 merge) ----
__global__ __launch_bounds__(192)
void k_rowfin(void* wsraw, int t) {
  WS w = carve(wsraw);
  const int r = threadIdx.x;
  if (r >= BK) return;
  const int k = r % KK;
  const int fl = w.flagA[r];
  const float sc = w.score[r];
  const float ln = w.len[r];

  if (fl) {  // finished beam: constant new_score = sc/denom_f -> min only
    atomicMin(&w.minbits[t], f2mono(sc / powf(ln + 1e-9f, LPEN)));
    return;
  }
  const float denom = powf(ln + 1.0f + 1e-9f, LPEN);

  float sm[SEGS];
  float rowmax = -__builtin_inff(), rowmin = __builtin_inff();
#pragma unroll
  for (int s = 0; s < SEGS; ++s) {
    sm[s] = w.seg_max[r * SEGS + s];
    rowmax = fmaxf(rowmax, sm[s]);
    rowmin = fminf(rowmin, w.seg_min[r * SEGS + s]);
  }
  float Z = 0.0f;
#pragma unroll
  for (int s = 0; s < SEGS; ++s)
    Z += w.seg_sum[r * SEGS + s] * expf(sm[s] - rowmax);
  const float logZ = logf(Z);

  const float candmin = sc + ((rowmin - rowmax) - logZ);
  atomicMin(&w.minbits[t], f2mono(candmin / denom));

  // 4-way merge of sorted per-segment top-10s -> row top-10
  int p[SEGS] = {0, 0, 0, 0};
  for (int j = 0; j < 10; ++j) {
    float bv = -__builtin_inff(); int bi = SENT_IDX; int bs = -1;
#pragma unroll
    for (int s = 0; s < SEGS; ++s) {
      if (p[s] < 10) {
        const float tv = w.seg_tv[(r * SEGS + s) * 10 + p[s]];
        const int   ti = w.seg_ti[(r * SEGS + s) * 10 + p[s]];
        if (bs < 0 || better(tv, ti, bv, bi)) { bv = tv; bi = ti; bs = s; }
      }
    }
#pragma unroll
    for (int s = 0; s < SEGS; ++s) if (s == bs) p[s]++;
    const float cand = sc + ((bv - rowmax) - logZ);
    w.part_val[r * 10 + j]  = cand / denom;
    w.part_cand[r * 10 + j] = cand;
    w.part_idx[r * 10 + j]  = k * VV + bi;
  }
}

// ---- kernel B: per-batch merge, exact top-10 (tie-break), state update ----
__global__ __launch_bounds__(128)
void k_merge(void* wsraw, const int* __restrict__ eosp, int t) {
  WS w = carve(wsraw);
  const int b = blockIdx.x;
  const int tid = threadIdx.x;
  const int eos = eosp[0];

  __shared__ int   oflag[KK];
  __shared__ float oscore[KK], olen[KK];
  __shared__ float cv[128], cc[128];
  __shared__ int   ci[128];
  __shared__ float rv[128];
  __shared__ int   ri[128];
  __shared__ int   selidx[10];
  __shared__ float selcand[10];

  if (tid < KK) {
    oflag[tid]  = w.flagA[b * KK + tid];
    oscore[tid] = w.score[b * KK + tid];
    olen[tid]   = w.len[b * KK + tid];
  }
  __syncthreads();

  const float gmin = mono2f(w.minbits[t]);

  float val = -__builtin_inff(); int idx = SENT_IDX; float cnd = 0.0f;
  if (tid < 100) {                       // active-beam partials
    const int kk = tid / 10, j = tid % 10;
    if (!oflag[kk]) {
      const int o = (b * KK + kk) * 10 + j;
      val = w.part_val[o]; idx = w.part_idx[o]; cnd = w.part_cand[o];
    }
  } else if (tid < 110) {                // finished beam, v=0 (unpenalized)
    const int kk = tid - 100;
    if (oflag[kk]) {
      const float df = powf(olen[kk] + 1e-9f, LPEN);
      val = oscore[kk] / df; idx = kk * VV; cnd = oscore[kk];
    }
  } else if (tid < 120) {                // finished beam, best penalized entry (v=1)
    const int kk = tid - 110;
    if (oflag[kk]) {
      const float df = powf(olen[kk] + 1e-9f, LPEN);
      val = oscore[kk] / df + 10.0f * gmin; idx = kk * VV + 1; cnd = oscore[kk];
    }
  }
  cv[tid] = val; ci[tid] = idx; cc[tid] = cnd;
  __syncthreads();

  for (int j = 0; j < 10; ++j) {
    rv[tid] = cv[tid]; ri[tid] = ci[tid]; __syncthreads();
    for (int s = 64; s > 0; s >>= 1) {
      if (tid < s && better(rv[tid + s], ri[tid + s], rv[tid], ri[tid])) {
        rv[tid] = rv[tid + s]; ri[tid] = ri[tid + s];
      }
      __syncthreads();
    }
    const int bidx = ri[0];
    if (tid == 0) selidx[j] = bidx;
    if (ci[tid] == bidx) {               // unique owner removes itself
      selcand[j] = cc[tid];
      cv[tid] = -__builtin_inff(); ci[tid] = SENT_IDX;
    }
    __syncthreads();
  }

  if (tid < KK) {
    const int idx2 = selidx[tid];
    const int rg = idx2 / VV;
    const int wid = idx2 - rg * VV;
    const int em = oflag[rg];                 // eosmet = gathered old flag
    const int validj = oflag[tid] ? 0 : 1;    // reference uses slot-j valid here
    const int nf = (em | ((wid == eos) && validj)) ? 1 : 0;
    const float nl = olen[rg] + (float)validj;
    const int o = t * BK + b * KK + tid;
    w.rec_wid[o] = wid; w.rec_rg[o] = rg; w.rec_em[o] = em;
    w.flagA[b * KK + tid] = nf;
    w.score[b * KK + tid] = selcand[tid];
    w.len[b * KK + tid]   = nl;
  }
}

// ---------------- backtracking + output ----------------
__global__ void k_backtrack(void* wsraw, float* __restrict__ out) {
  WS w = carve(wsraw);
  const int tid = threadIdx.x;
  if (tid >= BK) return;
  const int b = tid / KK, j = tid % KK;
  int idx = j, lensum = 0;
  for (int t = TT - 1; t >= 0; --t) {
    const int o = t * BK + b * KK + idx;
    const int wid = w.rec_wid[o];
    const int em = w.rec_em[o];
    const int keep = 1 - em;
    out[t * BK + b * KK + j] = (float)(wid * keep);
    lensum += keep;
    idx = w.rec_rg[o];
  }
  out[TT * BK + tid]      = (float)lensum;   // length [B,K]
  out[TT * BK + BK + tid] = w.score[tid];    // score  [B,K]
}

// ---------------- host entry ----------------
extern "C" void kernel_launch(void* const* d_in, const int* in_sizes, int n_in,
                              void* d_out, int out_size, void* d_ws, size_t ws_size,
                              hipStream_t stream) {
  (void)in_sizes; (void)n_in; (void)out_size; (void)ws_size;
  const float* logits = (const float*)d_in[0];
  const int*   eosp   = (const int*)d_in[1];
  k_setup<<<1, 256, 0, stream>>>(d_ws);
  for (int t = 0; t < TT; ++t) {
    k_seg<<<BK * SEGS, TPB_A, 0, stream>>>(logits, d_ws, t);
    k_rowfin<<<1, 192, 0, stream>>>(d_ws, t);
    k_merge<<<BB, 128, 0, stream>>>(d_ws, eosp, t);
  }
  k_backtrack<<<1, 192, 0, stream>>>(d_ws, (float*)d_out);
}